// Handcrafted_68135361183867
// MI455X (gfx1250) — compile-verified
//
#include <hip/hip_runtime.h>
#include <cstdint>

#define IMG_H 1024
#define IMG_W 1024
#define IMG_B 16
#define TILE  32
#define NTHREADS 256

// LDS tile geometry (odd strides -> no bank conflicts)
#define SX_W 36
#define SX_S 37
#define SD_W 34
#define SD_S 35

typedef float v2f __attribute__((ext_vector_type(2)));

// ---------- CDNA5 async global->LDS path (probe via __has_builtin) ----------
#if defined(__has_builtin)
#  if __has_builtin(__builtin_amdgcn_global_load_async_to_lds_b32)
#    define ASYNC_LDS_BUILTIN 1
#  endif
#  if __has_builtin(__builtin_amdgcn_s_wait_asynccnt)
#    define HAVE_WAIT_ASYNCCNT 1
#  endif
#endif

typedef __attribute__((address_space(1))) int g_i32;
typedef __attribute__((address_space(3))) int l_i32;

__device__ __forceinline__ void async_copy_f32(const float* __restrict__ g,
                                               float* __restrict__ l) {
#if defined(ASYNC_LDS_BUILTIN)
  // global src, LDS dst, imm offset, cpol
  __builtin_amdgcn_global_load_async_to_lds_b32(
      (g_i32*)(uintptr_t)g,
      (l_i32*)(unsigned)(uintptr_t)l,
      0, 0);
#else
  *l = *g;  // fallback: plain load + ds_store
#endif
}

__device__ __forceinline__ void async_wait_all() {
#if defined(ASYNC_LDS_BUILTIN)
#  if defined(HAVE_WAIT_ASYNCCNT)
  __builtin_amdgcn_s_wait_asynccnt(0);
#  else
  asm volatile("s_wait_asynccnt 0" ::: "memory");
#  endif
#endif
}

// ---------------------------------------------------------------------------
// Diagnostic-only (never launched): minimal body so the disasm snippet shows
// exactly what the async global->LDS builtin lowers to on gfx1250.
__global__ void cdna5_async_probe_kernel(const float* __restrict__ g,
                                         float* __restrict__ out) {
  __shared__ float buf[32];
  async_copy_f32(g + threadIdx.x, &buf[threadIdx.x]);
  async_wait_all();
  __syncthreads();
  out[threadIdx.x] = buf[threadIdx.x ^ 1];
}

// ---------------------------------------------------------------------------
// Shared tile pipeline. PASS 0: reduce 7 primitive abs-maxima into gmax[0..6]
// (fp bit patterns, all >= 0). PASS 1: recompute features, scale, write NHWC.
template <int PASS>
__device__ __forceinline__ void sobel_tile(const float* __restrict__ x,
                                           float* __restrict__ out,
                                           unsigned* __restrict__ gmax,
                                           const float* __restrict__ scales) {
  const int tid = threadIdx.x;
  const int ox0 = blockIdx.x * TILE;
  const int oy0 = blockIdx.y * TILE;
  const int b   = blockIdx.z;
  const float* xb = x + (size_t)b * IMG_H * IMG_W;

  __shared__ float sx [SX_W * SX_S];   // x tile with 2-px halo   (36x36)
  __shared__ float sdx[SD_W * SD_S];   // dx with 1-px halo        (34x34)
  __shared__ float sdy[SD_W * SD_S];   // dy with 1-px halo
  __shared__ float sred[64];           // 8 waves x 8 slots

  // ---- Stage 1: stage x halo tile into LDS via async DMA -------------------
  for (int i = tid; i < SX_W * SX_W; i += NTHREADS) {
    const int ly = i / SX_W, lx = i - ly * SX_W;
    const int gy = oy0 - 2 + ly, gx = ox0 - 2 + lx;
    float* dst = &sx[ly * SX_S + lx];
    if ((unsigned)gy < IMG_H && (unsigned)gx < IMG_W)
      async_copy_f32(xb + (size_t)gy * IMG_W + gx, dst);
    else
      *dst = 0.0f;                     // zero padding of x
  }
  async_wait_all();
  __syncthreads();

  // ---- Stage 2: first-order Sobel (cross-correlation, XLA convention) ------
  // dx/dy are zero OUTSIDE the image (that's what the 2nd conv's SAME pad sees)
  for (int i = tid; i < SD_W * SD_W; i += NTHREADS) {
    const int ly = i / SD_W, lx = i - ly * SD_W;
    const int gy = oy0 - 1 + ly, gx = ox0 - 1 + lx;
    float vdx = 0.0f, vdy = 0.0f;
    if ((unsigned)gy < IMG_H && (unsigned)gx < IMG_W) {
      const float* c = &sx[(ly + 1) * SX_S + (lx + 1)];
      const float a00 = c[-SX_S - 1], a01 = c[-SX_S], a02 = c[-SX_S + 1];
      const float a10 = c[-1],                         a12 = c[+1];
      const float a20 = c[ SX_S - 1], a21 = c[ SX_S], a22 = c[ SX_S + 1];
      vdx = (a02 - a00) + 2.0f * (a12 - a10) + (a22 - a20);
      vdy = (a20 - a00) + 2.0f * (a21 - a01) + (a22 - a02);
    }
    sdx[ly * SD_S + lx] = vdx;
    sdy[ly * SD_S + lx] = vdy;
  }
  __syncthreads();

  // ---- Stage 3: second-order stencils + products ---------------------------
  float m[7] = {0, 0, 0, 0, 0, 0, 0};
  float s[10];
  if (PASS == 1) {
#pragma unroll
    for (int k = 0; k < 10; ++k) s[k] = scales[k];  // uniform, precomputed
  }

  for (int i = tid; i < TILE * TILE; i += NTHREADS) {
    const int ly = i / TILE, lx = i - ly * TILE;
    const float* cx = &sdx[(ly + 1) * SD_S + (lx + 1)];
    const float* cy = &sdy[(ly + 1) * SD_S + (lx + 1)];
    const float dx = cx[0], dy = cy[0];
    const float dxx = (cx[-SD_S + 1] - cx[-SD_S - 1]) +
                      2.0f * (cx[1] - cx[-1]) +
                      (cx[SD_S + 1] - cx[SD_S - 1]);
    const float dyy = (cy[SD_S - 1] + 2.0f * cy[SD_S] + cy[SD_S + 1]) -
                      (cy[-SD_S - 1] + 2.0f * cy[-SD_S] + cy[-SD_S + 1]);
    const float dxy = (cx[SD_S - 1] + 2.0f * cx[SD_S] + cx[SD_S + 1]) -
                      (cx[-SD_S - 1] + 2.0f * cx[-SD_S] + cx[-SD_S + 1]);
    const float dxdy = dx * dy, dxxdyy = dxx * dyy;

    if (PASS == 0) {
      m[0] = fmaxf(m[0], fabsf(dx));
      m[1] = fmaxf(m[1], fabsf(dxx));
      m[2] = fmaxf(m[2], fabsf(dy));
      m[3] = fmaxf(m[3], fabsf(dyy));
      m[4] = fmaxf(m[4], fabsf(dxy));
      m[5] = fmaxf(m[5], fabsf(dxdy));
      m[6] = fmaxf(m[6], fabsf(dxxdyy));
    } else {
      const int gy = oy0 + ly, gx = ox0 + lx;
      float* o = out + ((size_t)(b * IMG_H + gy) * IMG_W + gx) * 10;
      const v2f p0 = {dx * s[0],      dx * dx * s[1]};
      const v2f p1 = {dxx * s[2],     dy * s[3]};
      const v2f p2 = {dy * dy * s[4], dyy * s[5]};
      const v2f p3 = {dxdy * s[6],    dxxdyy * s[7]};
      const v2f p4 = {dxy * s[8],     dxy * dxy * s[9]};
      __builtin_nontemporal_store(p0, (v2f*)(o + 0));   // 40B stride: 8B aligned
      __builtin_nontemporal_store(p1, (v2f*)(o + 2));
      __builtin_nontemporal_store(p2, (v2f*)(o + 4));
      __builtin_nontemporal_store(p3, (v2f*)(o + 6));
      __builtin_nontemporal_store(p4, (v2f*)(o + 8));
    }
  }

  if (PASS == 0) {
    // wave32 butterfly reduce, then cross-wave via LDS, then 7 global atomics
#pragma unroll
    for (int k = 0; k < 7; ++k) {
      float v = m[k];
#pragma unroll
      for (int d = 16; d >= 1; d >>= 1)
        v = fmaxf(v, __shfl_xor(v, d, 32));
      m[k] = v;
    }
    const int wave = tid >> 5, lane = tid & 31;
    if (lane == 0) {
#pragma unroll
      for (int k = 0; k < 7; ++k) sred[wave * 8 + k] = m[k];
    }
    __syncthreads();
    if (tid < 7) {
      float v = 0.0f;
#pragma unroll
      for (int w = 0; w < 8; ++w) v = fmaxf(v, sred[w * 8 + tid]);
      atomicMax(&gmax[tid], __float_as_uint(v));  // valid: v >= 0
    }
  }
}

__global__ __launch_bounds__(NTHREADS) void sobel_pass0_kernel(
    const float* __restrict__ x, unsigned* __restrict__ gmax) {
  sobel_tile<0>(x, nullptr, gmax, nullptr);
}

__global__ __launch_bounds__(NTHREADS) void sobel_pass1_kernel(
    const float* __restrict__ x, float* __restrict__ out,
    const float* __restrict__ scales) {
  sobel_tile<1>(x, out, nullptr, scales);
}

// Turn the 7 abs-maxima into the 10 per-feature inverse scales.
__global__ void sobel_scale_kernel(const unsigned* __restrict__ gmax,
                                   float* __restrict__ scales) {
  if (threadIdx.x == 0) {
    const float mdx   = __uint_as_float(gmax[0]);
    const float mdxx  = __uint_as_float(gmax[1]);
    const float mdy   = __uint_as_float(gmax[2]);
    const float mdyy  = __uint_as_float(gmax[3]);
    const float mdxy  = __uint_as_float(gmax[4]);
    const float mdxdy = __uint_as_float(gmax[5]);
    const float mxxyy = __uint_as_float(gmax[6]);
    scales[0] = 1.0f / mdx;   scales[1] = 1.0f / (mdx * mdx);
    scales[2] = 1.0f / mdxx;  scales[3] = 1.0f / mdy;
    scales[4] = 1.0f / (mdy * mdy);  scales[5] = 1.0f / mdyy;
    scales[6] = 1.0f / mdxdy; scales[7] = 1.0f / mxxyy;
    scales[8] = 1.0f / mdxy;  scales[9] = 1.0f / (mdxy * mdxy);
  }
}

__global__ void sobel_init_kernel(unsigned* gmax) {
  if (threadIdx.x < 8) gmax[threadIdx.x] = 0u;
}

extern "C" void kernel_launch(void* const* d_in, const int* in_sizes, int n_in,
                              void* d_out, int out_size, void* d_ws, size_t ws_size,
                              hipStream_t stream) {
  (void)in_sizes; (void)n_in; (void)out_size; (void)ws_size;
  const float* x = (const float*)d_in[0];
  float* out = (float*)d_out;
  unsigned* gmax = (unsigned*)d_ws;            // ws[0..7]  : abs-max bits
  float* scales = (float*)d_ws + 8;            // ws[8..17] : inverse scales

  dim3 grid(IMG_W / TILE, IMG_H / TILE, IMG_B);
  sobel_init_kernel<<<1, 32, 0, stream>>>(gmax);
  sobel_pass0_kernel<<<grid, NTHREADS, 0, stream>>>(x, gmax);
  sobel_scale_kernel<<<1, 32, 0, stream>>>(gmax, scales);
  sobel_pass1_kernel<<<grid, NTHREADS, 0, stream>>>(x, out, scales);
}